// LIF_2714419331191
// MI455X (gfx1250) — compile-verified
//
#include <hip/hip_runtime.h>

// LIF hard-reset scan, B=16, S=256, H=128, N=64, T=S*H=32768.
// d_in[0]=inputs[B,S,H] f32, d_in[1]=threshes[N] f32, d_in[2]=acc0[B,N] f32.
// d_out = outs[B,S,N,H] f32 ++ spikes[B,S,N,H] f32 (spikes emitted as 0/1).
//
// Store-bound op: 268 MB of output stores -> ~11.5us floor @ 23.3 TB/s.
// Two passes: (1) exact sequential state sweep at row granularity,
// (2) massively parallel replay+emit using the CDNA5 async LDS engine.

#define Bq 16
#define Sq 256
#define Hq 128
#define Nq 64
#define PADR 68   // 64 data floats + 4 pad: rows stay 16B-aligned, <=2-way bank conflict

// ---------------- CDNA5 async global<->LDS path (guarded) ----------------
#if defined(__has_builtin)
#if __has_builtin(__builtin_amdgcn_global_load_async_to_lds_b128) && \
    __has_builtin(__builtin_amdgcn_global_store_async_from_lds_b128) && \
    __has_builtin(__builtin_amdgcn_s_wait_asynccnt)
#define USE_ASYNC_LDS 1
#endif
#endif
#ifndef USE_ASYNC_LDS
#define USE_ASYNC_LDS 0
#endif

#if USE_ASYNC_LDS
#define GAS __attribute__((address_space(1)))
#define LAS __attribute__((address_space(3)))
typedef int v4i __attribute__((ext_vector_type(4)));
__device__ __forceinline__ void ldsmov_load_b128(const float* g, float* l) {
  // GLOBAL_LOAD_ASYNC_TO_LDS_B128: per-lane 16B, tracked by ASYNCcnt
  __builtin_amdgcn_global_load_async_to_lds_b128((GAS v4i*)g, (LAS v4i*)l, 0, 0);
}
__device__ __forceinline__ void ldsmov_store_b128(float* g, const float* l) {
  // GLOBAL_STORE_ASYNC_FROM_LDS_B128: per-lane 16B, no VGPR data
  __builtin_amdgcn_global_store_async_from_lds_b128((GAS v4i*)g, (LAS v4i*)l, 0, 0);
}
__device__ __forceinline__ void wait_async() { __builtin_amdgcn_s_wait_asynccnt(0); }
#else
__device__ __forceinline__ void ldsmov_load_b128(const float* g, float* l) {
  *(float4*)l = *(const float4*)g;
}
__device__ __forceinline__ void ldsmov_store_b128(float* g, const float* l) {
  *(float4*)g = *(const float4*)l;
}
__device__ __forceinline__ void wait_async() {}
#endif

// One LIF step (exact reference semantics: strict >, reset to 0 on fire)
__device__ __forceinline__ void lif_step(float x, float t, float& acc, float& o, float& sp) {
  acc += x;
  const bool f = acc > t;
  o   = f ? acc : 0.f;
  sp  = f ? 1.f : 0.f;
  acc = f ? 0.f : acc;
}

// ---------------- Pass 1: exact sequential sweep, record row-entry states ----------------
// 16 blocks (one per batch) x 32 lanes; lane owns chains n=lane and n=lane+32
// (2-way ILP to hide the 3-op dependent chain; VOPD can dual-issue the pair).
// Entry state of every s-row is written into the spikes region at (b,s,n,h=0);
// pass 2 reads exactly those slots from the region it alone rewrites (race-free).
__global__ __launch_bounds__(32) void lif_states(const float* __restrict__ in,
                                                 const float* __restrict__ th,
                                                 const float* __restrict__ acc0,
                                                 float* __restrict__ stateBase) {
  const int b    = blockIdx.x;
  const int lane = threadIdx.x;                 // 0..31, single wave
  __shared__ __align__(16) float xrow[2][Hq];   // double-buffered input row

  float a0 = acc0[b * Nq + lane];
  float a1 = acc0[b * Nq + lane + 32];
  const float t0 = th[lane];
  const float t1 = th[lane + 32];
  const size_t inBase = (size_t)b * Sq * Hq;

  // prefetch row 0: 32 lanes x 16B = whole 512B row in one async op
  ldsmov_load_b128(&in[inBase + lane * 4], &xrow[0][lane * 4]);
  wait_async();
  __syncthreads();

  for (int s = 0; s < Sq; ++s) {
    const int cur = s & 1;
    if (s + 1 < Sq)
      ldsmov_load_b128(&in[inBase + (size_t)(s + 1) * Hq + lane * 4],
                       &xrow[cur ^ 1][lane * 4]);

    // record row-entry states (consumed by the emit pass)
    const size_t slot = (size_t)(b * Sq + s) * Nq;
    stateBase[(slot + lane)      * Hq] = a0;
    stateBase[(slot + lane + 32) * Hq] = a1;

    #pragma unroll 4
    for (int h = 0; h < Hq; h += 4) {
      const float4 xv = *(const float4*)&xrow[cur][h];
      a0 += xv.x; a0 = (a0 > t0) ? 0.f : a0;
      a1 += xv.x; a1 = (a1 > t1) ? 0.f : a1;
      a0 += xv.y; a0 = (a0 > t0) ? 0.f : a0;
      a1 += xv.y; a1 = (a1 > t1) ? 0.f : a1;
      a0 += xv.z; a0 = (a0 > t0) ? 0.f : a0;
      a1 += xv.z; a1 = (a1 > t1) ? 0.f : a1;
      a0 += xv.w; a0 = (a0 > t0) ? 0.f : a0;
      a1 += xv.w; a1 = (a1 > t1) ? 0.f : a1;
    }
    wait_async();        // next-row buffer ready
    __syncthreads();     // single-wave WG: barrier degenerates to S_NOP
  }
}

// ---------------- Pass 2: parallel replay + coalesced async emit ----------------
// 4096 blocks (one per (b,s) row) x 64 threads (one per neuron), 2 waves.
// Replays 128 steps from the recorded entry state into a padded LDS tile,
// then emits the 32KB row fully line-coalesced with async LDS->global B128.
__global__ __launch_bounds__(64) void lif_emit(const float* __restrict__ in,
                                               const float* __restrict__ th,
                                               float* __restrict__ outBase,
                                               float* __restrict__ spkBase) {
  const int bs = blockIdx.x;          // b*S + s
  const int n  = threadIdx.x;         // 0..63
  __shared__ __align__(16) float xrow[Hq];
  __shared__ __align__(16) float obuf[Nq * PADR];
  __shared__ __align__(16) float sbuf[Nq * PADR];

  const size_t rowBase = (size_t)bs * (Nq * Hq);

  if (n < 32) {                       // wave 0 stages the 512B input row
    ldsmov_load_b128(&in[(size_t)bs * Hq + n * 4], &xrow[n * 4]);
    wait_async();
  }
  float acc     = spkBase[rowBase + (size_t)n * Hq];  // entry state from pass 1
  const float t = th[n];
  __syncthreads();

  for (int half = 0; half < 2; ++half) {
    // compute 64 steps into the transposed LDS tile
    #pragma unroll 4
    for (int hq = 0; hq < 16; ++hq) {
      const float4 xv = *(const float4*)&xrow[half * 64 + hq * 4];
      const int base = n * PADR + hq * 4;
      float o, sp;
      lif_step(xv.x, t, acc, o, sp); obuf[base + 0] = o; sbuf[base + 0] = sp;
      lif_step(xv.y, t, acc, o, sp); obuf[base + 1] = o; sbuf[base + 1] = sp;
      lif_step(xv.z, t, acc, o, sp); obuf[base + 2] = o; sbuf[base + 2] = sp;
      lif_step(xv.w, t, acc, o, sp); obuf[base + 3] = o; sbuf[base + 3] = sp;
    }
    __syncthreads();

    // cooperative emit: 1024 B128 chunks per array, 16 per thread, coalesced
    #pragma unroll
    for (int k = 0; k < 16; ++k) {
      const int c  = n + 64 * k;       // chunk id 0..1023
      const int rn = c >> 4;           // neuron row
      const int h4 = (c & 15) * 4;     // h offset within half
      const size_t g = rowBase + (size_t)rn * Hq + half * 64 + h4;
      ldsmov_store_b128(&outBase[g], &obuf[rn * PADR + h4]);
      ldsmov_store_b128(&spkBase[g], &sbuf[rn * PADR + h4]);
    }
    wait_async();        // LDS tile may be reused / WG may exit
    __syncthreads();
  }
}

extern "C" void kernel_launch(void* const* d_in, const int* in_sizes, int n_in,
                              void* d_out, int out_size, void* d_ws, size_t ws_size,
                              hipStream_t stream) {
  (void)in_sizes; (void)n_in; (void)out_size; (void)d_ws; (void)ws_size;
  const float* inp  = (const float*)d_in[0];
  const float* th   = (const float*)d_in[1];
  const float* acc0 = (const float*)d_in[2];
  float* out = (float*)d_out;
  float* spk = out + (size_t)Bq * Sq * Nq * Hq;   // second output region

  // Pass 1: exact sequential state sweep (states stashed in spikes region)
  lif_states<<<dim3(Bq), dim3(32), 0, stream>>>(inp, th, acc0, spk);
  // Pass 2: parallel replay + bandwidth-bound emit (overwrites all of d_out)
  lif_emit<<<dim3(Bq * Sq), dim3(64), 0, stream>>>(inp, th, out, spk);
}